// PocStrengthNet_31885837205794
// MI455X (gfx1250) — compile-verified
//
#include <hip/hip_runtime.h>
#include <math.h>

typedef __attribute__((ext_vector_type(2))) float v2f;
typedef __attribute__((ext_vector_type(4))) float v4f;
typedef __attribute__((ext_vector_type(8))) float v8f;

#define XS  68   // padded LDS row stride for x tile (floats); 272B = 16B multiple
#define WPS 96   // pair-row stride for interleaved W1 (floats); 96 mod 64 = 32 -> halves on disjoint banks

// monotone float <-> uint key for atomic max over signed floats
__device__ __forceinline__ unsigned fkey(float x) {
    unsigned u = __float_as_uint(x);
    return (u & 0x80000000u) ? ~u : (u | 0x80000000u);
}
__device__ __forceinline__ float funkey(unsigned k) {
    unsigned u = (k & 0x80000000u) ? (k ^ 0x80000000u) : ~k;
    return __uint_as_float(u);
}

__global__ __launch_bounds__(256) void k_init(unsigned* __restrict__ segmax_u,
                                              float* __restrict__ denom,
                                              float* __restrict__ numer, int S) {
    int i = blockIdx.x * 256 + threadIdx.x;
    if (i < S) { segmax_u[i] = 0u; denom[i] = 0.0f; numer[i] = 0.0f; }
}

// 8 waves/block, each wave computes h = relu(x[16x64] @ W1[64x32] + b1) via WMMA f32,
// then r = h.wr + br, z = h.wz + bz, written to workspace.
__global__ __launch_bounds__(256) void k_gemm_heads(
    const float* __restrict__ x, const float* __restrict__ W1,
    const float* __restrict__ b1, const float* __restrict__ wr,
    const float* __restrict__ brp, const float* __restrict__ wz,
    const float* __restrict__ bzp,
    float* __restrict__ r_out, float* __restrict__ z_out, int N)
{
    // W1 pair-interleaved: sW1p[p*WPS + 2*h + s] = W1[2p+s][h], p=0..31, h=0..31, s=0..1
    __shared__ __align__(16) float sW1p[32 * WPS];  // 12 KB
    __shared__ __align__(16) float sX[8][16 * XS];  // ~34.8 KB

    const int tid  = threadIdx.x;
    const int wave = tid >> 5;
    const int lane = tid & 31;

    // W1 (64x32 row-major) -> pair-interleaved LDS, coalesced global reads
    #pragma unroll
    for (int i0 = 0; i0 < 8; ++i0) {
        int i  = i0 * 256 + tid;        // 0..2047
        int kk = i >> 5, hh = i & 31;
        sW1p[(kk >> 1) * WPS + 2 * hh + (kk & 1)] = W1[i];
    }

    // this wave's 16x64 f32 tile -> LDS via b128, coalesced
    const int rowBase = (blockIdx.x * 8 + wave) * 16;
    {
        const v4f* xg = (const v4f*)(x + (size_t)rowBase * 64);
        #pragma unroll
        for (int i = 0; i < 8; ++i) {
            int idx = i * 32 + lane;      // 0..255
            int row = idx >> 4;
            int c4  = idx & 15;
            *(v4f*)&sX[wave][row * XS + c4 * 4] = xg[row * 16 + c4];
        }
    }
    // prefetch a future block's tile into cache hierarchy (global_prefetch_b8)
    if (rowBase + 2048 < N) __builtin_prefetch(&x[(size_t)(rowBase + 2048) * 64], 0, 1);

    __syncthreads();

    const int   n    = lane & 15;
    const int   koff = (lane >> 4) << 1;   // 0 for lanes 0-15, 2 for lanes 16-31
    const float b1lo = b1[n],      b1hi = b1[16 + n];
    const float wrlo = wr[n],      wrhi = wr[16 + n];
    const float wzlo = wz[n],      wzhi = wz[16 + n];
    const float br   = brp[0],     bz   = bzp[0];

    v8f c0 = {}; v8f c1 = {};
    const float* xrow = &sX[wave][n * XS];                       // A: M = lane%16
    const float* wrow = &sW1p[(lane >> 4) * WPS + 2 * n];        // lanes>=16 start one pair later

    #pragma unroll
    for (int k = 0; k < 64; k += 4) {
        // A fragment 16x4: v0 = K{k+koff}, v1 = K{k+koff+1}  (ds_load_b64, conflict-free)
        v2f a  = *(const v2f*)&xrow[k + koff];
        // B fragments 4x16: single ds_load_b64 each, lands in adjacent VGPR pair
        v2f bA = *(const v2f*)&wrow[(k >> 1) * WPS];        // columns 0..15
        v2f bB = *(const v2f*)&wrow[(k >> 1) * WPS + 32];   // columns 16..31
        c0 = __builtin_amdgcn_wmma_f32_16x16x4_f32(false, a, false, bA, (short)0, c0, false, false);
        c1 = __builtin_amdgcn_wmma_f32_16x16x4_f32(false, a, false, bB, (short)0, c1, false, false);
    }

    // heads: per accumulator VGPR v, lanes 0-15 hold row v, lanes 16-31 hold row v+8
    #pragma unroll
    for (int v = 0; v < 8; ++v) {
        float h0 = fmaxf(c0[v] + b1lo, 0.0f);   // h columns 0..15  (col = lane%16)
        float h1 = fmaxf(c1[v] + b1hi, 0.0f);   // h columns 16..31
        float rp = fmaf(h0, wrlo, h1 * wrhi);
        float zp = fmaf(h0, wzlo, h1 * wzhi);
        #pragma unroll
        for (int s = 1; s < 16; s <<= 1) {      // butterfly within 16-lane halves
            rp += __shfl_xor(rp, s, 32);
            zp += __shfl_xor(zp, s, 32);
        }
        if ((lane & 15) == 0) {
            int row = rowBase + v + ((lane >> 4) << 3);
            r_out[row] = rp + br;
            z_out[row] = zp + bz;
        }
    }
}

// sorted ids: segmented max within each wave's contiguous 32 rows, one atomic per run head
__global__ __launch_bounds__(256) void k_segmax(
    const float* __restrict__ z, const int* __restrict__ ids,
    unsigned* __restrict__ segmax_u, int N)
{
    int i = blockIdx.x * 256 + threadIdx.x;
    if (i >= N) return;
    int lane = threadIdx.x & 31;
    int sid  = ids[i];
    float m  = z[i];
    #pragma unroll
    for (int s = 1; s < 32; s <<= 1) {
        float om = __shfl_down(m, s, 32);
        int   os = __shfl_down(sid, s, 32);
        if (lane + s < 32 && os == sid) m = fmaxf(m, om);
    }
    int up = __shfl_up(sid, 1, 32);
    if (lane == 0 || up != sid) atomicMax(&segmax_u[sid], fkey(m));
}

__global__ __launch_bounds__(256) void k_segsum(
    const float* __restrict__ z, const float* __restrict__ r,
    const int* __restrict__ ids, const unsigned* __restrict__ segmax_u,
    float* __restrict__ denom, float* __restrict__ numer, int N)
{
    int i = blockIdx.x * 256 + threadIdx.x;
    if (i >= N) return;
    int lane = threadIdx.x & 31;
    int sid  = ids[i];
    float m   = funkey(segmax_u[sid]);
    float ez  = __expf(z[i] - m);
    float ezr = ez * r[i];
    #pragma unroll
    for (int s = 1; s < 32; s <<= 1) {
        float o1 = __shfl_down(ez,  s, 32);
        float o2 = __shfl_down(ezr, s, 32);
        int   os = __shfl_down(sid, s, 32);
        if (lane + s < 32 && os == sid) { ez += o1; ezr += o2; }
    }
    int up = __shfl_up(sid, 1, 32);
    if (lane == 0 || up != sid) {
        atomicAdd(&denom[sid], ez);
        atomicAdd(&numer[sid], ezr);
    }
}

__global__ __launch_bounds__(256) void k_final(
    const float* __restrict__ denom, const float* __restrict__ numer,
    float* __restrict__ out, int S)
{
    int i = blockIdx.x * 256 + threadIdx.x;
    if (i < S) {
        float d = denom[i];
        float p = (d > 0.0f) ? (numer[i] / d) : 7.6699353278706015f;
        out[i] = 173.71779276130213f * p;   // 400/ln(10)
    }
}

extern "C" void kernel_launch(void* const* d_in, const int* in_sizes, int n_in,
                              void* d_out, int out_size, void* d_ws, size_t ws_size,
                              hipStream_t stream)
{
    const float* x   = (const float*)d_in[0];
    const float* W1  = (const float*)d_in[1];
    const float* b1  = (const float*)d_in[2];
    const float* wr  = (const float*)d_in[3];
    const float* br  = (const float*)d_in[4];
    const float* wz  = (const float*)d_in[5];
    const float* bz  = (const float*)d_in[6];
    const int*   ids = (const int*)d_in[7];

    const int N = in_sizes[0] / 64;     // rows of x
    const int S = out_size;             // num_segments

    float*    r_ws     = (float*)d_ws;
    float*    z_ws     = r_ws + N;
    unsigned* segmax_u = (unsigned*)(z_ws + N);
    float*    denom    = (float*)(segmax_u + S);
    float*    numer    = denom + S;

    k_init      <<<(S + 255) / 256, 256, 0, stream>>>(segmax_u, denom, numer, S);
    k_gemm_heads<<<N / 128,          256, 0, stream>>>(x, W1, b1, wr, br, wz, bz, r_ws, z_ws, N);
    k_segmax    <<<(N + 255) / 256,  256, 0, stream>>>(z_ws, ids, segmax_u, N);
    k_segsum    <<<(N + 255) / 256,  256, 0, stream>>>(z_ws, r_ws, ids, segmax_u, denom, numer, N);
    k_final     <<<(S + 255) / 256,  256, 0, stream>>>(denom, numer, (float*)d_out, S);
}